// QualityContrastiveLoss_81166291960034
// MI455X (gfx1250) — compile-verified
//
#include <hip/hip_runtime.h>
#include <hip/hip_bf16.h>
#include <math.h>

typedef __attribute__((ext_vector_type(2))) float v2f;
typedef __attribute__((ext_vector_type(8))) float v8f;

#define TEMPERATURE 0.07f
#define INV_T (1.0f / TEMPERATURE)
#define QUAL_TH 0.1f
#define NEG_TH 0.3f
#define TOPK 10
#define EPS_L 1e-8f

// ---------------------------------------------------------------------------
// Kernel 0: zero the two global accumulators (total, count)
// ---------------------------------------------------------------------------
__global__ void init_acc_kernel(float* acc) {
    acc[0] = 0.0f;
    acc[1] = 0.0f;
}

// ---------------------------------------------------------------------------
// Kernel 1: L2-normalize each row (norm clamped at 1e-12, like the reference)
// blockDim.x == 256, one block per row, D <= 256
// ---------------------------------------------------------------------------
__global__ __launch_bounds__(256)
void normalize_kernel(const float* __restrict__ f, float* __restrict__ fnorm, int D) {
    const int row = blockIdx.x;
    const int t = threadIdx.x;
    const float* src = f + (size_t)row * D;

    float v = (t < D) ? src[t] : 0.0f;
    __shared__ float red[256];
    red[t] = v * v;
    __syncthreads();
    #pragma unroll
    for (int off = 128; off > 0; off >>= 1) {
        if (t < off) red[t] += red[t + off];
        __syncthreads();
    }
    const float nrm = sqrtf(red[0]);
    const float scale = 1.0f / fmaxf(nrm, 1e-12f);
    if (t < D) fnorm[(size_t)row * D + t] = v * scale;
}

// ---------------------------------------------------------------------------
// Pass kernel (templated):
//   PASS1: per-row top-10 negative sims -> sum_exp_neg, valid flags
//   PASS2: accumulate positive-pair loss using sum_exp_neg
// Workgroup = 16 rows, 256 threads = 8 waves. Column chunks of 128 (= 8 waves
// x one 16x16 WMMA tile each). K = D (256) via v_wmma_f32_16x16x4_f32.
// ---------------------------------------------------------------------------
template <bool PASS1>
__global__ __launch_bounds__(256)
void sim_pass_kernel(const float* __restrict__ fnorm,
                     const float* __restrict__ q,
                     float* __restrict__ sumexp,
                     int* __restrict__ validArr,
                     float* __restrict__ acc,
                     int Bn, int Dd) {
    const int t = threadIdx.x;
    const int i0 = blockIdx.x * 16;

    __shared__ float sA[16][256];        // 16 A rows (this block's rows)
    __shared__ float sSim[16][128];      // sim chunk (scaled by 1/T)
    __shared__ float sQj[128];           // quality of column chunk
    __shared__ float sQi[16];            // quality of our 16 rows
    __shared__ int   sNpos[16], sNneg[16];
    __shared__ float sMerge[16][16][10]; // pass1 top-k merge buffer
    __shared__ float sRed[256];          // pass2 block reduction
    __shared__ float sCi[16];
    __shared__ int   sValid[16];

    // Stage A rows into LDS
    for (int idx = t; idx < 16 * 256; idx += 256) {
        const int rr = idx >> 8, kk = idx & 255;
        sA[rr][kk] = (kk < Dd) ? fnorm[(size_t)(i0 + rr) * Dd + kk] : 0.0f;
    }
    if (t < 16) {
        sQi[t] = q[i0 + t];
        sNpos[t] = 0;
        sNneg[t] = 0;
        if (!PASS1) {
            sCi[t] = sumexp[i0 + t];
            sValid[t] = validArr[i0 + t];
        }
    }
    __syncthreads();

    // WMMA operand mapping (wave32)
    const int wave = t >> 5;             // 0..7 -> column sub-tile
    const int lane = t & 31;
    const int m = lane & 15;             // row (A) / col (B) within 16
    const int kh = (lane >> 4) << 1;     // K sub-pair: 0 or 2
    const int rowoff = (lane >> 4) << 3; // C rows: lanes>=16 hold M+8

    // Scan-phase mapping: 16 row-teams x 16 threads
    const int r = t >> 4;
    const int tt = t & 15;
    const int irow = i0 + r;
    const float qir = sQi[r];
    const float Ci = PASS1 ? 0.0f : sCi[r];
    const int rvalid = PASS1 ? 1 : sValid[r];

    float topv[TOPK];
    #pragma unroll
    for (int p = 0; p < TOPK; ++p) topv[p] = -INFINITY;
    int npos = 0, nneg = 0;
    float lsum = 0.0f, lcnt = 0.0f;

    for (int c0 = 0; c0 < Bn; c0 += 128) {
        if (t < 128) sQj[t] = q[c0 + t];

        // ---- WMMA phase: 16x16 fp32 tile, K = Dd ----
        const int jc = c0 + (wave << 4);
        v8f cacc = {};
        const float* arow = &sA[m][kh];
        const float* brow = fnorm + (size_t)(jc + m) * Dd + kh;
        for (int k = 0; k < Dd; k += 4) {
            v2f a = *(const v2f*)(arow + k);
            v2f b = *(const v2f*)(brow + k);
            cacc = __builtin_amdgcn_wmma_f32_16x16x4_f32(
                false, a, false, b, (short)0, cacc, false, false);
        }
        #pragma unroll
        for (int rr = 0; rr < 8; ++rr)
            sSim[rr + rowoff][(wave << 4) + m] = cacc[rr] * INV_T;
        __syncthreads();

        // ---- Scan phase ----
        if (PASS1) {
            for (int cc = tt; cc < 128; cc += 16) {
                const int j = c0 + cc;
                const float s = sSim[r][cc];
                const float qd = fabsf(qir - sQj[cc]);
                if (j != irow) {
                    if (qd > NEG_TH) {
                        ++nneg;
                        if (s > topv[TOPK - 1]) {
                            float x = s;
                            #pragma unroll
                            for (int p = 0; p < TOPK; ++p) {
                                const float vp = topv[p];
                                if (x > vp) { topv[p] = x; x = vp; }
                            }
                        }
                    } else if (qd < QUAL_TH) {
                        ++npos;
                    }
                }
            }
        } else {
            if (rvalid) {
                for (int cc = tt; cc < 128; cc += 16) {
                    const int j = c0 + cc;
                    const float s = sSim[r][cc];
                    const float qd = fabsf(qir - sQj[cc]);
                    if (j != irow && qd < QUAL_TH) {
                        lsum += logf(expf(s) + Ci + EPS_L) - s;
                        lcnt += 1.0f;
                    }
                }
            }
        }
        __syncthreads();
    }

    if (PASS1) {
        // Merge 16 per-thread top-10 lists per row -> true top-10
        #pragma unroll
        for (int p = 0; p < TOPK; ++p) sMerge[r][tt][p] = topv[p];
        atomicAdd(&sNpos[r], npos);
        atomicAdd(&sNneg[r], nneg);
        __syncthreads();
        if (t < 16) {
            float w[TOPK];
            #pragma unroll
            for (int p = 0; p < TOPK; ++p) w[p] = -INFINITY;
            for (int src = 0; src < 16; ++src) {
                #pragma unroll
                for (int p = 0; p < TOPK; ++p) {
                    float x = sMerge[t][src][p];
                    if (x > w[TOPK - 1]) {
                        #pragma unroll
                        for (int pp = 0; pp < TOPK; ++pp) {
                            const float vp = w[pp];
                            if (x > vp) { w[pp] = x; x = vp; }
                        }
                    }
                }
            }
            float se = 0.0f;
            #pragma unroll
            for (int p = 0; p < TOPK; ++p)
                if (w[p] > -INFINITY) se += expf(w[p]); // -inf pad -> exp 0
            sumexp[i0 + t] = se;
            validArr[i0 + t] = (sNpos[t] > 0 && sNneg[t] > 0) ? 1 : 0;
        }
    } else {
        // Block-reduce loss sum and pair count, then global atomic add
        sRed[t] = lsum;
        __syncthreads();
        #pragma unroll
        for (int off = 128; off > 0; off >>= 1) {
            if (t < off) sRed[t] += sRed[t + off];
            __syncthreads();
        }
        if (t == 0) atomicAdd(&acc[0], sRed[0]);
        __syncthreads();
        sRed[t] = lcnt;
        __syncthreads();
        #pragma unroll
        for (int off = 128; off > 0; off >>= 1) {
            if (t < off) sRed[t] += sRed[t + off];
            __syncthreads();
        }
        if (t == 0) atomicAdd(&acc[1], sRed[0]);
    }
}

// ---------------------------------------------------------------------------
// Finalize: mean over positive pairs (0 if none)
// ---------------------------------------------------------------------------
__global__ void finalize_kernel(const float* acc, float* out) {
    const float total = acc[0];
    const float count = acc[1];
    out[0] = (count > 0.0f) ? (total / fmaxf(count, 1.0f)) : 0.0f;
}

// ---------------------------------------------------------------------------
extern "C" void kernel_launch(void* const* d_in, const int* in_sizes, int n_in,
                              void* d_out, int out_size, void* d_ws, size_t ws_size,
                              hipStream_t stream) {
    const float* feat = (const float*)d_in[0];   // [B, D] f32
    const float* q    = (const float*)d_in[1];   // [B]    f32
    float* out = (float*)d_out;

    const int Bn = in_sizes[1];
    const int Dd = in_sizes[0] / Bn;

    // Workspace layout
    float* fnorm   = (float*)d_ws;                       // B*D floats
    float* sumexp  = fnorm + (size_t)Bn * Dd;            // B floats
    int*   validA  = (int*)(sumexp + Bn);                // B ints
    float* acc     = (float*)(validA + Bn);              // 2 floats

    init_acc_kernel<<<1, 1, 0, stream>>>(acc);
    normalize_kernel<<<Bn, 256, 0, stream>>>(feat, fnorm, Dd);
    sim_pass_kernel<true ><<<Bn / 16, 256, 0, stream>>>(fnorm, q, sumexp, validA, acc, Bn, Dd);
    sim_pass_kernel<false><<<Bn / 16, 256, 0, stream>>>(fnorm, q, sumexp, validA, acc, Bn, Dd);
    finalize_kernel<<<1, 1, 0, stream>>>(acc, out);
}